// get_coordinate_77653008712115
// MI455X (gfx1250) — compile-verified
//
#include <hip/hip_runtime.h>
#include <stdint.h>

#define WG   256
#define CCH  16          // channels

// fused level1+2 tile: each block produces TY x TX c2 outputs
#define TY   4
#define TX   8
#define FR   (4*TY+3)    // 19 staged feat rows
#define FC   (4*TX+3)    // 35 staged feat cols
#define C1R  (2*TY+1)    // 9 c1 rows (incl. halo)
#define C1C  (2*TX+1)    // 17 c1 cols (incl. halo)

#if defined(__has_builtin)
#  if __has_builtin(__builtin_amdgcn_global_load_async_to_lds_b128)
#    define HAVE_ASYNC_B128 1
#  endif
#  if __has_builtin(__builtin_amdgcn_s_wait_asynccnt)
#    define HAVE_WAIT_ASYNC 1
#  endif
#endif

typedef int v4i __attribute__((ext_vector_type(4)));

// CDNA5 async global->LDS copy of 16 bytes (per active lane), ASYNCcnt-tracked.
__device__ __forceinline__ void cp16_g2l_async(const float* g, float* l) {
#if defined(HAVE_ASYNC_B128)
  typedef __attribute__((address_space(1))) v4i* gp_t;   // global v4i*
  typedef __attribute__((address_space(3))) v4i* lp_t;   // LDS v4i*
  __builtin_amdgcn_global_load_async_to_lds_b128(
      (gp_t)(uintptr_t)g,
      (lp_t)(uint32_t)(uintptr_t)l,   // low 32 bits of generic LDS ptr = LDS byte addr
      0, 0);
#else
  // VDST = VGPR holding LDS byte address; VADDR = 64-bit global address
  asm volatile("global_load_async_to_lds_b128 %0, %1, off"
               : : "v"((uint32_t)(uintptr_t)l), "v"(g) : "memory");
#endif
}

__device__ __forceinline__ void wait_async0() {
#if defined(HAVE_WAIT_ASYNC)
  __builtin_amdgcn_s_wait_asynccnt(0);
#else
  asm volatile("s_wait_asynccnt 0" : : : "memory");
#endif
}

// ---------------- level 1 + level 2 fused ----------------
__global__ void __launch_bounds__(WG)
fused_pool12(const float* __restrict__ feat,
             const unsigned char* __restrict__ mask,
             float* __restrict__ c2,
             int H, int W, int H2, int W2) {
  __shared__ __align__(16) float s_feat[FR*FC*CCH];   // 42,560 B
  __shared__ __align__(16) float s_c1[C1R*C1C*CCH];   //  9,792 B
  __shared__ float s_m[FR*FC];                        //  2,660 B

  const int tid = threadIdx.x;
  const int X0 = blockIdx.x * TX;    // c2 tile origin
  const int Y0 = blockIdx.y * TY;
  const int b  = blockIdx.z;
  const int r0 = 4*Y0 - 3;           // feat row of LDS row 0
  const int c0 = 4*X0 - 3;           // feat col of LDS col 0

  // ---- stage mask tile as floats (0 outside the image => handles padding) ----
  for (int n = tid; n < FR*FC; n += WG) {
    const int ly = n / FC, lx = n - ly*FC;
    const int gy = r0 + ly, gx = c0 + lx;
    float v = 0.0f;
    if (gy >= 0 && gy < H && gx >= 0 && gx < W)
      v = mask[((size_t)b*H + gy)*(size_t)W + gx] ? 1.0f : 0.0f;
    s_m[n] = v;
  }

  // ---- async-stage feat tile (clamped addresses; OOB taps killed by s_m==0) ----
  const int NCHUNK = FR*FC*4;                 // float4 chunks = 2660
  const int NITER  = (NCHUNK + WG - 1) / WG;  // 11
  for (int it = 0; it < NITER; ++it) {
    int n = it*WG + tid;
    n = (n < NCHUNK) ? n : (NCHUNK - 1);      // keep all lanes active (dup copy is benign)
    const int coord = n >> 2;
    const int q     = n & 3;
    const int ly = coord / FC, lx = coord - ly*FC;
    int gy = r0 + ly; gy = gy < 0 ? 0 : (gy > H-1 ? H-1 : gy);
    int gx = c0 + lx; gx = gx < 0 ? 0 : (gx > W-1 ? W-1 : gx);
    const float* src = feat + (((size_t)b*H + gy)*(size_t)W + gx)*CCH + q*4;
    cp16_g2l_async(src, &s_feat[coord*CCH + q*4]);
  }
  wait_async0();
  __syncthreads();

  // ---- level 1: compute c1 halo tile (masked input, m1 gating) into LDS ----
  for (int t = tid; t < C1R*C1C; t += WG) {
    const int i = t / C1C, j = t - i*C1C;     // c1 row = 2*Y0-1+i, col = 2*X0-1+j
    float acc[CCH];
#pragma unroll
    for (int k = 0; k < CCH; ++k) acc[k] = 0.0f;
#pragma unroll
    for (int dy = 0; dy < 3; ++dy) {
#pragma unroll
      for (int dx = 0; dx < 3; ++dx) {
        const int fy = 2*i + dy, fx = 2*j + dx;
        const float w = s_m[fy*FC + fx];
        const float* p = &s_feat[(fy*FC + fx)*CCH];
#pragma unroll
        for (int k = 0; k < CCH; ++k) acc[k] += w * p[k];
      }
    }
    // m1 = any mask over the 2x2 strided block {2c, 2c+1}^2 (LDS rows 2i+1,2i+2)
    const float occ = s_m[(2*i+1)*FC + (2*j+1)] + s_m[(2*i+1)*FC + (2*j+2)]
                    + s_m[(2*i+2)*FC + (2*j+1)] + s_m[(2*i+2)*FC + (2*j+2)];
    const float keep = occ > 0.0f ? 1.0f : 0.0f;
    float* o = &s_c1[t*CCH];
#pragma unroll
    for (int k = 0; k < CCH; ++k) o[k] = keep * acc[k];
  }
  __syncthreads();

  // ---- level 2: 4x8x16 c2 tile from the c1 LDS tile, m2 gating ----
  if (tid < TY*TX*4) {
    const int q   = tid & 3;
    const int idx = tid >> 2;
    const int oy = idx / TX, ox = idx - oy*TX;
    float a0 = 0.f, a1 = 0.f, a2 = 0.f, a3 = 0.f;
#pragma unroll
    for (int dy = 0; dy < 3; ++dy) {
#pragma unroll
      for (int dx = 0; dx < 3; ++dx) {
        const float* p = &s_c1[((2*oy+dy)*C1C + (2*ox+dx))*CCH + q*4];
        a0 += p[0]; a1 += p[1]; a2 += p[2]; a3 += p[3];
      }
    }
    // m2 = any mask over the 4x4 block rows 4*y2..4*y2+3 (LDS rows 4*oy+3..4*oy+6)
    float occ = 0.0f;
#pragma unroll
    for (int yy = 0; yy < 4; ++yy)
#pragma unroll
      for (int xx = 0; xx < 4; ++xx)
        occ += s_m[(4*oy+3+yy)*FC + (4*ox+3+xx)];
    const float keep = occ > 0.0f ? 1.0f : 0.0f;
    const int y2 = Y0 + oy, x2 = X0 + ox;
    float4 r; r.x = keep*a0; r.y = keep*a1; r.z = keep*a2; r.w = keep*a3;
    *(float4*)(c2 + (((size_t)b*H2 + y2)*(size_t)W2 + x2)*CCH + q*4) = r;
  }
}

// ---------------- level 3 ----------------
__global__ void __launch_bounds__(WG)
pool3(const float* __restrict__ c2,
      const unsigned char* __restrict__ mask,
      float* __restrict__ c3,
      int B, int H3, int W3, int H2, int W2, int Hm, int Wm) {
  const int g = blockIdx.x * WG + threadIdx.x;
  const int total = B * H3 * W3 * 4;
  if (g >= total) return;
  const int q   = g & 3;
  const int idx = g >> 2;
  const int x3 = idx % W3;
  const int y3 = (idx / W3) % H3;
  const int b  = idx / (W3 * H3);

  float a0 = 0.f, a1 = 0.f, a2 = 0.f, a3 = 0.f;
#pragma unroll
  for (int dy = -1; dy <= 1; ++dy) {
    const int r = 2*y3 + dy;
    if (r < 0 || r >= H2) continue;
#pragma unroll
    for (int dx = -1; dx <= 1; ++dx) {
      const int c = 2*x3 + dx;
      if (c < 0 || c >= W2) continue;
      const float4 v = *(const float4*)(c2 + (((size_t)b*H2 + r)*(size_t)W2 + c)*CCH + q*4);
      a0 += v.x; a1 += v.y; a2 += v.z; a3 += v.w;
    }
  }
  // m3 = any mask over the 8x8 block (8 aligned 8-byte row loads, OR-reduced)
  unsigned long long occ = 0ull;
  const unsigned char* mb = mask + ((size_t)b*Hm + (size_t)y3*8)*(size_t)Wm + (size_t)x3*8;
#pragma unroll
  for (int ry = 0; ry < 8; ++ry)
    occ |= *(const unsigned long long*)(mb + (size_t)ry*Wm);
  const float keep = occ ? 1.0f : 0.0f;
  float4 r4; r4.x = keep*a0; r4.y = keep*a1; r4.z = keep*a2; r4.w = keep*a3;
  *(float4*)(c3 + (size_t)idx*CCH + q*4) = r4;
}

extern "C" void kernel_launch(void* const* d_in, const int* in_sizes, int n_in,
                              void* d_out, int out_size, void* d_ws, size_t ws_size,
                              hipStream_t stream) {
  (void)in_sizes; (void)n_in; (void)d_ws; (void)ws_size; (void)out_size;
  const float*         feat = (const float*)d_in[0];
  const unsigned char* mask = (const unsigned char*)d_in[1];  // jax bool -> 1 byte
  const int B = 4, H = 1024, W = 1024;
  const int H2 = H/4, W2 = W/4;   // 256
  const int H3 = H/8, W3 = W/8;   // 128

  float* c2 = (float*)d_out;                               // (4,256,256,16)
  float* c3 = c2 + (size_t)B*H2*W2*CCH;                    // (4,128,128,16)

  dim3 grid1(W2/TX, H2/TY, B);   // (32, 64, 4)
  fused_pool12<<<grid1, dim3(WG), 0, stream>>>(feat, mask, c2, H, W, H2, W2);

  const int total3 = B*H3*W3*4;
  pool3<<<dim3((total3 + WG - 1)/WG), dim3(WG), 0, stream>>>(
      c2, mask, c3, B, H3, W3, H2, W2, H, W);
}